// DigitCapsules_37031208026161
// MI455X (gfx1250) — compile-verified
//
#include <hip/hip_runtime.h>
#include <math.h>

// DigitCapsules dynamic routing, fused single kernel for gfx1250 (MI455X).
// x:   [B=128, N=1152, Ci=8]  f32
// w:   [D=10, N=1152, Ci=8, Co=16] f32
// out: [D=10, B=128, 1, 1, Co=16] f32
//
// Per block: one (d, 16-row batch tile). u[n] = x_tile(16x8) @ w[d,n](8x16)
// computed with V_WMMA_F32_16X16X4_F32 (K=8 -> 2 WMMAs, no padding waste).
// Routing (3 iters) fused; u recomputed per pass from L2-resident operands
// (x: 4.7 MB + w: 5.9 MB << 192 MB L2; avoids 94 MB of HBM traffic for u).

typedef __attribute__((ext_vector_type(2))) float v2f;
typedef __attribute__((ext_vector_type(8))) float v8f;

#define NPRIM  1152
#define DOUT   16
#define CIN    8
#define BTILE  16
#define NWAVES 8
#define NPW    (NPRIM / NWAVES)   // 144 n per wave, uniform trip count

__device__ __forceinline__ v8f wmma_u(v8f cin,
                                      const float* __restrict__ Xb,
                                      const float* __restrict__ Wd,
                                      int n, int half, int lo) {
  // A frag: 16x4 f32. lane<16 -> K{0,1}, lane>=16 -> K{2,3}; second WMMA: +4.
  const float* xp = Xb + (size_t)lo * (NPRIM * CIN) + (size_t)n * CIN + half * 2;
  v2f a0; a0.x = xp[0]; a0.y = xp[1];
  v2f a1; a1.x = xp[4]; a1.y = xp[5];
  // B frag: 4x16 f32, same K split; w[d,n,k,o], k-stride = 16 floats.
  const float* wp = Wd + (size_t)n * (CIN * DOUT) + half * 2 * DOUT + lo;
  v2f b0; b0.x = wp[0];        b0.y = wp[DOUT];
  v2f b1; b1.x = wp[4 * DOUT]; b1.y = wp[5 * DOUT];
  v8f u = __builtin_amdgcn_wmma_f32_16x16x4_f32(
      false, a0, false, b0, (short)0, cin, false, false);
  u = __builtin_amdgcn_wmma_f32_16x16x4_f32(
      false, a1, false, b1, (short)0, u, false, false);
  return u;
}

__global__ __launch_bounds__(256)
void digit_caps_routing_kernel(const float* __restrict__ X,
                               const float* __restrict__ W,
                               float* __restrict__ Out) {
  __shared__ float lds_logits[NPRIM][BTILE];     // 73728 B, raw routing logits
  __shared__ float lds_red[NWAVES][BTILE][DOUT]; // cross-wave s reduction
  __shared__ float lds_v[BTILE][DOUT];           // squashed capsule vector
  __shared__ float lds_pmax[16][BTILE];
  __shared__ float lds_psum[16][BTILE];
  __shared__ float lds_gmax[BTILE];
  __shared__ float lds_ginv[BTILE];

  const int tid  = threadIdx.x;
  const int lane = tid & 31;
  const int wv   = tid >> 5;     // wave id 0..7, owns n = wv + 8k (uniform)
  const int half = lane >> 4;    // 0: b in [0,8), 1: b in [8,16)
  const int lo   = lane & 15;
  const int d    = blockIdx.y;
  const int b0   = blockIdx.x * BTILE;

  const float* Xb = X + (size_t)b0 * NPRIM * CIN;
  const float* Wd = W + (size_t)d * NPRIM * CIN * DOUT;

  const int rb = tid >> 4;       // b index for 256-thread (b,o) mapping
  const int ro = tid & 15;       // o index
  const v8f vzero = {};

  // zero routing logits once so the logit pass is always "+=" (no branch)
  for (int i = tid; i < NPRIM * BTILE; i += 256)
    ((float*)lds_logits)[i] = 0.0f;

  // ---------- Pass A: s0 = mean_n u (uniform softmax of zero logits) ----------
  v8f acc = vzero;
#pragma unroll 2
  for (int k = 0; k < NPW; ++k)
    acc = wmma_u(acc, Xb, Wd, wv + k * NWAVES, half, lo);
#pragma unroll
  for (int r = 0; r < 8; ++r) lds_red[wv][r + half * 8][lo] = acc[r];
  __syncthreads();
  {
    float s = 0.f;
#pragma unroll
    for (int q = 0; q < NWAVES; ++q) s += lds_red[q][rb][ro];
    s *= (1.0f / NPRIM);
    // elementwise squash (faithful to reference's singleton-axis norm)
    lds_v[rb][ro] = s * fabsf(s) / (1.0f + s * s);
  }
  __syncthreads();

  for (int it = 0; it < 2; ++it) {
    // ---------- Pass B/D: logits[n][b] += sum_o u[n,b,o] * v[b,o] ----------
    for (int k = 0; k < NPW; ++k) {
      const int n = wv + k * NWAVES;
      v8f u = wmma_u(vzero, Xb, Wd, n, half, lo);
#pragma unroll
      for (int r = 0; r < 8; ++r) {
        const int bb = r + half * 8;
        float p = u[r] * lds_v[bb][lo];
        p += __shfl_xor(p, 1);   // butterfly sum over 16 o-lanes (wave32 halves)
        p += __shfl_xor(p, 2);
        p += __shfl_xor(p, 4);
        p += __shfl_xor(p, 8);
        if (lo == 0) lds_logits[n][bb] += p;
      }
    }
    __syncthreads();

    // ---------- softmax over n: per-b max and inverse-sum ----------
    {
      const int sb = tid & 15, part = tid >> 4;
      float m = -3.402823466e38f;
      for (int n = part; n < NPRIM; n += 16) m = fmaxf(m, lds_logits[n][sb]);
      lds_pmax[part][sb] = m;
      __syncthreads();
      if (tid < 16) {
        float mm = lds_pmax[0][tid];
#pragma unroll
        for (int q = 1; q < 16; ++q) mm = fmaxf(mm, lds_pmax[q][tid]);
        lds_gmax[tid] = mm;
      }
      __syncthreads();
      const float gm = lds_gmax[sb];
      float sum = 0.f;
      for (int n = part; n < NPRIM; n += 16)
        sum += __expf(lds_logits[n][sb] - gm);
      lds_psum[part][sb] = sum;
      __syncthreads();
      if (tid < 16) {
        float t = 0.f;
#pragma unroll
        for (int q = 0; q < 16; ++q) t += lds_psum[q][tid];
        lds_ginv[tid] = 1.0f / t;
      }
      __syncthreads();
    }

    // ---------- Pass C/E: s = sum_n c[n,b] * u[n,b,o] ----------
    v8f acc2 = vzero;
#pragma unroll 2
    for (int k = 0; k < NPW; ++k) {
      const int n = wv + k * NWAVES;
      v8f u = wmma_u(vzero, Xb, Wd, n, half, lo);
#pragma unroll
      for (int r = 0; r < 8; ++r) {
        const int bb = r + half * 8;
        const float cw =
            __expf(lds_logits[n][bb] - lds_gmax[bb]) * lds_ginv[bb];
        acc2[r] += cw * u[r];
      }
    }
#pragma unroll
    for (int r = 0; r < 8; ++r) lds_red[wv][r + half * 8][lo] = acc2[r];
    __syncthreads();
    {
      float s = 0.f;
#pragma unroll
      for (int q = 0; q < NWAVES; ++q) s += lds_red[q][rb][ro];
      const float vv = s * fabsf(s) / (1.0f + s * s);
      if (it == 1) {
        Out[((size_t)d * 128 + (b0 + rb)) * DOUT + ro] = vv;
      } else {
        lds_v[rb][ro] = vv;
      }
    }
    __syncthreads();
  }
}

extern "C" void kernel_launch(void* const* d_in, const int* in_sizes, int n_in,
                              void* d_out, int out_size, void* d_ws, size_t ws_size,
                              hipStream_t stream) {
  (void)in_sizes; (void)n_in; (void)out_size; (void)d_ws; (void)ws_size;
  const float* x = (const float*)d_in[0];   // [128,1152,8]
  const float* w = (const float*)d_in[1];   // [10,1152,8,16]
  float* out = (float*)d_out;               // [10,128,1,1,16]
  dim3 grid(128 / BTILE, 10, 1);            // (b-tile, d)
  digit_caps_routing_kernel<<<grid, dim3(256, 1, 1), 0, stream>>>(x, w, out);
}